// MemoryBank_15642270892501
// MI455X (gfx1250) — compile-verified
//
#include <hip/hip_runtime.h>
#include <math.h>

typedef float v2f __attribute__((ext_vector_type(2)));
typedef float v8f __attribute__((ext_vector_type(8)));

#define NQ 4096
#define D 128
#define NM 100000
#define QT 128                 // queries per workgroup
#define QTILES (NQ / QT)       // 32
#define SLABS 25
#define SLAB_ROWS (NM / SLABS) // 4000
#define TILES (SLAB_ROWS / 16) // 250
#define NEG_INF (-3.402823466e38f)

// ---------------- Kernel 1: normalize queries ----------------
__global__ __launch_bounds__(128) void qnorm_kernel(const float* __restrict__ q,
                                                    float* __restrict__ qn) {
  int row = blockIdx.x;
  int t = threadIdx.x;
  float v = q[row * D + t];
  __shared__ float red[128];
  red[t] = v * v;
  __syncthreads();
  for (int s = 64; s > 0; s >>= 1) {
    if (t < s) red[t] += red[t + s];
    __syncthreads();
  }
  float inv = 1.0f / fmaxf(sqrtf(red[0]), 1e-12f);
  qn[row * D + t] = v * inv;
}

// ---------------- Kernel 2: fused normalize(mem) + WMMA sims + per-slab top-3 ----------------
__global__ __launch_bounds__(256) void topk_slab_kernel(const float* __restrict__ qn,
                                                        const float* __restrict__ mem,
                                                        float* __restrict__ cand_v,
                                                        int* __restrict__ cand_i) {
  __shared__ float smem_m[16 * D];     // normalized memory tile (16 rows x 128)
  __shared__ float partials[16][16];   // row-norm partial sums
  __shared__ float sinv[16];           // per-row inverse norms
  __shared__ float candv[8 * 32 * 8 * 3];
  __shared__ int   candi[8 * 32 * 8 * 3];

  const int qtile = blockIdx.x;   // 0..31
  const int slab  = blockIdx.y;   // 0..24
  const int tid   = threadIdx.x;  // 0..255
  const int wave  = tid >> 5;     // 0..7
  const int lane  = tid & 31;
  const int lane15 = lane & 15;
  const int laneh  = lane >> 4;   // 0 or 1

  // ---- Load A fragments (this wave's 16 normalized query rows) into registers.
  // ISA 32-bit A 16x4 layout: lanes 0-15 hold (M=lane, K=k0,k0+1), lanes 16-31 hold (M=lane-16, K=k0+2,k0+3)
  const int qrow = qtile * QT + wave * 16 + lane15;
  const float* qp = qn + qrow * D + laneh * 2;
  v2f aF[32];
#pragma unroll
  for (int j = 0; j < 32; ++j) {
    aF[j] = *reinterpret_cast<const v2f*>(qp + j * 4);
  }

  // ---- Per-lane running top-3 for 8 C row-slots (rows m = slot + 8*laneh), col class n = lane15
  float tv0[8], tv1[8], tv2[8];
  int   ti0[8], ti1[8], ti2[8];
#pragma unroll
  for (int i = 0; i < 8; ++i) {
    tv0[i] = tv1[i] = tv2[i] = NEG_INF;
    ti0[i] = ti1[i] = ti2[i] = 0;
  }

  const int slab_base = slab * SLAB_ROWS;

  for (int t = 0; t < TILES; ++t) {
    const int rowbase = slab_base + t * 16;

    // Cooperative load: 16 rows x 128 f32 = 2048 floats; 8 per thread (two float4)
    const float4* g = reinterpret_cast<const float4*>(mem + rowbase * D);
    float4 r0 = g[tid * 2 + 0];
    float4 r1 = g[tid * 2 + 1];
    // Row-norm partial from registers (thread's 8 floats are all in row tid>>4)
    float ss = r0.x * r0.x + r0.y * r0.y + r0.z * r0.z + r0.w * r0.w +
               r1.x * r1.x + r1.y * r1.y + r1.z * r1.z + r1.w * r1.w;
    partials[tid >> 4][tid & 15] = ss;
    __syncthreads();

    if (tid < 16) {
      float s = 0.0f;
#pragma unroll
      for (int i = 0; i < 16; ++i) s += partials[tid][i];
      sinv[tid] = 1.0f / fmaxf(sqrtf(s), 1e-12f);
    }
    __syncthreads();

    // Scale in registers, stage normalized tile to LDS
    float sc = sinv[tid >> 4];
    r0.x *= sc; r0.y *= sc; r0.z *= sc; r0.w *= sc;
    r1.x *= sc; r1.y *= sc; r1.z *= sc; r1.w *= sc;
    float4* sm = reinterpret_cast<float4*>(smem_m);
    sm[tid * 2 + 0] = r0;
    sm[tid * 2 + 1] = r1;
    __syncthreads();

    // ---- 16x16 fp32 C tile via 32 chained v_wmma_f32_16x16x4_f32 (K=128)
    // B (4x16, K x N): lane holds column N=lane15, K = k0 + 2*laneh + {0,1} (mirrors A layout)
    v8f c = {};
    const v2f* bp = reinterpret_cast<const v2f*>(smem_m + lane15 * D + laneh * 2);
#pragma unroll
    for (int j = 0; j < 32; ++j) {
      v2f b = bp[j * 2];
      c = __builtin_amdgcn_wmma_f32_16x16x4_f32(false, aF[j], false, b,
                                                (short)0, c, false, false);
    }

    // ---- Running top-3 update. C layout: VGPR i, lane l -> M = i + 8*(l>=16), N = l&15
    const int col = rowbase + lane15;
#pragma unroll
    for (int i = 0; i < 8; ++i) {
      float v = c[i];
      if (v > tv0[i]) {
        tv2[i] = tv1[i]; ti2[i] = ti1[i];
        tv1[i] = tv0[i]; ti1[i] = ti0[i];
        tv0[i] = v;      ti0[i] = col;
      } else if (v > tv1[i]) {
        tv2[i] = tv1[i]; ti2[i] = ti1[i];
        tv1[i] = v;      ti1[i] = col;
      } else if (v > tv2[i]) {
        tv2[i] = v;      ti2[i] = col;
      }
    }
    // No extra barrier needed: next iteration only touches smem_m after two
    // __syncthreads(), which all threads reach only after finishing frag reads.
  }

  // ---- Merge per-lane top-3 across the 16 column-class lanes of each row
  const int base = ((wave * 32 + lane) * 8) * 3;
#pragma unroll
  for (int i = 0; i < 8; ++i) {
    candv[base + i * 3 + 0] = tv0[i]; candi[base + i * 3 + 0] = ti0[i];
    candv[base + i * 3 + 1] = tv1[i]; candi[base + i * 3 + 1] = ti1[i];
    candv[base + i * 3 + 2] = tv2[i]; candi[base + i * 3 + 2] = ti2[i];
  }
  __syncthreads();

  if (tid < QT) {
    const int w = tid >> 4;        // wave that owned this query
    const int m = tid & 15;        // row within wave's 16-query tile
    const int hi = m >> 3;         // lane half that holds this row
    const int slot = m & 7;        // C VGPR slot
    float b0 = NEG_INF, b1 = NEG_INF, b2 = NEG_INF;
    int j0 = 0, j1 = 0, j2 = 0;
    for (int n = 0; n < 16; ++n) {
      const int l = hi * 16 + n;
      const int cb = ((w * 32 + l) * 8 + slot) * 3;
#pragma unroll
      for (int j = 0; j < 3; ++j) {
        float v = candv[cb + j];
        int ix = candi[cb + j];
        if (v > b0) {
          b2 = b1; j2 = j1; b1 = b0; j1 = j0; b0 = v; j0 = ix;
        } else if (v > b1) {
          b2 = b1; j2 = j1; b1 = v; j1 = ix;
        } else if (v > b2) {
          b2 = v; j2 = ix;
        }
      }
    }
    const int q = qtile * QT + tid;
    const int ob = (q * SLABS + slab) * 3;
    cand_v[ob + 0] = b0; cand_i[ob + 0] = j0;
    cand_v[ob + 1] = b1; cand_i[ob + 1] = j1;
    cand_v[ob + 2] = b2; cand_i[ob + 2] = j2;
  }
}

// ---------------- Kernel 3: merge per-slab candidates -> final top-3 ----------------
__global__ __launch_bounds__(256) void merge_kernel(const float* __restrict__ cand_v,
                                                    const int* __restrict__ cand_i,
                                                    float* __restrict__ out_d,
                                                    int* __restrict__ out_i) {
  const int q = blockIdx.x * blockDim.x + threadIdx.x;
  if (q >= NQ) return;
  const float* cv = cand_v + q * (SLABS * 3);
  const int* ci = cand_i + q * (SLABS * 3);
  float b0 = NEG_INF, b1 = NEG_INF, b2 = NEG_INF;
  int j0 = 0, j1 = 0, j2 = 0;
  for (int j = 0; j < SLABS * 3; ++j) {
    float v = cv[j];
    int ix = ci[j];
    if (v > b0) {
      b2 = b1; j2 = j1; b1 = b0; j1 = j0; b0 = v; j0 = ix;
    } else if (v > b1) {
      b2 = b1; j2 = j1; b1 = v; j1 = ix;
    } else if (v > b2) {
      b2 = v; j2 = ix;
    }
  }
  out_d[q * 3 + 0] = 1.0f - b0;
  out_d[q * 3 + 1] = 1.0f - b1;
  out_d[q * 3 + 2] = 1.0f - b2;
  out_i[q * 3 + 0] = j0;
  out_i[q * 3 + 1] = j1;
  out_i[q * 3 + 2] = j2;
}

extern "C" void kernel_launch(void* const* d_in, const int* in_sizes, int n_in,
                              void* d_out, int out_size, void* d_ws, size_t ws_size,
                              hipStream_t stream) {
  const float* queries = (const float*)d_in[0]; // [4096,128]
  const float* memory  = (const float*)d_in[1]; // [100000,128]
  // d_in[2] = k (always 3 per reference)

  // Workspace layout (needs ~4.6 MB):
  //   qn      : 4096*128 floats (normalized queries)
  //   cand_v  : 4096*25*3 floats (per-slab top-3 sims)
  //   cand_i  : 4096*25*3 ints  (per-slab top-3 indices)
  float* qn = (float*)d_ws;
  float* cand_v = qn + NQ * D;
  int* cand_i = (int*)(cand_v + NQ * SLABS * 3);

  qnorm_kernel<<<NQ, 128, 0, stream>>>(queries, qn);

  dim3 grid(QTILES, SLABS);
  topk_slab_kernel<<<grid, 256, 0, stream>>>(qn, memory, cand_v, cand_i);

  // d_out: [distances fp32 4096*3][indices int32 4096*3]
  float* out_d = (float*)d_out;
  int* out_i = (int*)d_out + NQ * 3;
  merge_kernel<<<NQ / 256, 256, 0, stream>>>(cand_v, cand_i, out_d, out_i);
}